// HeteroGCNLayer_10496900072192
// MI455X (gfx1250) — compile-verified
//
#include <hip/hip_runtime.h>
#include <hip/hip_bf16.h>

typedef __attribute__((ext_vector_type(2))) float v2f;
typedef __attribute__((ext_vector_type(8))) float v8f;

// ---------------------------------------------------------------------------
// GEMM: proj[mtile*16 .. +16, 0..128) = H_tile(16x64) @ W(64x128) + bias
// One block = 256 threads = 8 waves. Block handles one 16-row M tile.
// Wave w computes the 16x16 output tile for columns [w*16, w*16+16).
// K=64 swept as 16 chained V_WMMA_F32_16X16X4_F32 steps.
// Result written to both `proj` (scratch, read by edge gathers) and
// `outInit` (output buffer, which the scatter kernels accumulate into).
// ---------------------------------------------------------------------------
__global__ __launch_bounds__(256) void gemm16_wmma_kernel(
    const float* __restrict__ H,     // [nrows, 64] row-major
    const float* __restrict__ W,     // [64, 128] row-major
    const float* __restrict__ bias,  // [128]
    float* __restrict__ proj,        // [nrows, 128]
    float* __restrict__ outInit)     // [nrows, 128]
{
    __shared__ float Atile[16 * 64];

    const int mtile = blockIdx.x;
    const int tid   = threadIdx.x;

    // Stage the 16x64 f32 A tile in LDS (shared by all 8 waves).
    const float* __restrict__ Arow = H + (size_t)mtile * 16 * 64;
    #pragma unroll
    for (int i = tid; i < 16 * 64; i += 256) Atile[i] = Arow[i];
    __syncthreads();

    const int wave = tid >> 5;   // 0..7 -> N tile index
    const int lane = tid & 31;
    const int half = lane >> 4;  // 0: lanes 0-15, 1: lanes 16-31
    const int l16  = lane & 15;
    const int n    = wave * 16 + l16;   // output column for this lane

    v8f c = {};
    #pragma unroll
    for (int kk = 0; kk < 16; ++kk) {
        // A 16x4 fragment (2 VGPRs): lanes 0-15 hold K=kk*4+{0,1},
        // lanes 16-31 hold K=kk*4+{2,3}, M = lane%16 (ISA 7.12.2).
        const int k0 = kk * 4 + half * 2;
        v2f av, bv;
        av.x = Atile[l16 * 64 + k0];
        av.y = Atile[l16 * 64 + k0 + 1];
        // B 4x16 fragment: row K striped across lanes within a VGPR,
        // mirroring the A K-split across lane halves.
        bv.x = W[(size_t)k0 * 128 + n];
        bv.y = W[(size_t)(k0 + 1) * 128 + n];
        // 8 args: (neg_a, A, neg_b, B, c_mod, C, reuse_a, reuse_b)
        c = __builtin_amdgcn_wmma_f32_16x16x4_f32(
            false, av, false, bv, (short)0, c, false, false);
    }

    const float bval = bias[n];
    #pragma unroll
    for (int r = 0; r < 8; ++r) {
        // C/D layout: VGPR r, lanes 0-15 -> M=r, lanes 16-31 -> M=r+8.
        const size_t m = (size_t)mtile * 16 + half * 8 + r;
        const float v = c[r] + bval;
        proj[m * 128 + n]    = v;
        outInit[m * 128 + n] = v;
    }
}

// ---------------------------------------------------------------------------
// Zero-fill (degree counters must start at 0 every call).
// ---------------------------------------------------------------------------
__global__ void zero_kernel(float* __restrict__ p, size_t n) {
    size_t i = (size_t)blockIdx.x * blockDim.x + threadIdx.x;
    const size_t stride = (size_t)gridDim.x * blockDim.x;
    for (; i < n; i += stride) p[i] = 0.0f;
}

// ---------------------------------------------------------------------------
// Degree counting: deg[dst[e]] += 1  (clamped at use site).
// ---------------------------------------------------------------------------
__global__ void degree_kernel(const long long* __restrict__ dst, int ne,
                              float* __restrict__ deg) {
    int e = blockIdx.x * blockDim.x + threadIdx.x;
    if (e < ne) atomicAdd(&deg[dst[e]], 1.0f);
}

// ---------------------------------------------------------------------------
// op->op sequence edges: out_op[dst] += proj_op[src] / clamp(deg_seq[dst],1)
// One block per edge, 128 threads = 128 features (coalesced gather,
// non-returning f32 atomics into the L2-resident output).
// ---------------------------------------------------------------------------
__global__ __launch_bounds__(128) void scatter_seq_kernel(
    const long long* __restrict__ E, int ne,
    const float* __restrict__ proj_op,
    const float* __restrict__ deg_seq,
    float* __restrict__ out_op)
{
    const int e = blockIdx.x;
    const int f = threadIdx.x;
    const long long s = E[e];
    const long long d = E[(size_t)ne + e];
    const float inv = 1.0f / fmaxf(deg_seq[d], 1.0f);
    atomicAdd(&out_op[(size_t)d * 128 + f], proj_op[(size_t)s * 128 + f] * inv);
}

// ---------------------------------------------------------------------------
// op<->machine edges, both directions fused in one pass over 2M edges:
//   out_m[dst_m]  += proj_op[src_op] / clamp(deg_m[dst_m],1)
//   out_op[src_op]+= proj_m[dst_m]   / clamp(deg_op[src_op],1)
// Gathers hit the 51MB / 1MB proj tables (L2-resident on 192MB L2).
// ---------------------------------------------------------------------------
__global__ __launch_bounds__(128) void scatter_op2m_kernel(
    const long long* __restrict__ E, int ne,
    const float* __restrict__ proj_op,
    const float* __restrict__ proj_m,
    const float* __restrict__ deg_m,
    const float* __restrict__ deg_op,
    float* __restrict__ out_op,
    float* __restrict__ out_m)
{
    const int e = blockIdx.x;
    const int f = threadIdx.x;
    const long long s = E[e];                 // op index
    const long long d = E[(size_t)ne + e];    // machine index

    const float inv_m = 1.0f / fmaxf(deg_m[d], 1.0f);
    atomicAdd(&out_m[(size_t)d * 128 + f], proj_op[(size_t)s * 128 + f] * inv_m);

    const float inv_o = 1.0f / fmaxf(deg_op[s], 1.0f);
    atomicAdd(&out_op[(size_t)s * 128 + f], proj_m[(size_t)d * 128 + f] * inv_o);
}

// ---------------------------------------------------------------------------
// In-place ReLU over the whole concatenated output.
// ---------------------------------------------------------------------------
__global__ void relu_kernel(float* __restrict__ x, size_t n) {
    size_t i = (size_t)blockIdx.x * blockDim.x + threadIdx.x;
    const size_t stride = (size_t)gridDim.x * blockDim.x;
    for (; i < n; i += stride) x[i] = fmaxf(x[i], 0.0f);
}

extern "C" void kernel_launch(void* const* d_in, const int* in_sizes, int n_in,
                              void* d_out, int out_size, void* d_ws, size_t ws_size,
                              hipStream_t stream) {
    (void)n_in; (void)out_size; (void)ws_size;

    const float*     H_op   = (const float*)d_in[0];
    const float*     H_m    = (const float*)d_in[1];
    const long long* E_seq  = (const long long*)d_in[2];
    const long long* E_op2m = (const long long*)d_in[3];
    const float*     W_op   = (const float*)d_in[4];
    const float*     b_op   = (const float*)d_in[5];
    const float*     W_m    = (const float*)d_in[6];
    const float*     b_m    = (const float*)d_in[7];

    const int n_op  = in_sizes[0] / 64;   // 100000
    const int n_m   = in_sizes[1] / 64;   // 2000
    const int e_seq = in_sizes[2] / 2;    // 100000
    const int e_o2m = in_sizes[3] / 2;    // 2000000

    float* out_op = (float*)d_out;                      // [n_op,128]
    float* out_m  = out_op + (size_t)n_op * 128;        // [n_m,128]

    // Workspace layout (floats): proj_op | proj_m | deg_seq | deg_m | deg_op
    float* ws       = (float*)d_ws;
    float* proj_op  = ws;
    float* proj_m   = proj_op + (size_t)n_op * 128;
    float* deg_seq  = proj_m + (size_t)n_m * 128;
    float* deg_m    = deg_seq + n_op;
    float* deg_op   = deg_m + n_m;

    // 1) Reset degree counters (proj/out are fully overwritten below).
    zero_kernel<<<1024, 256, 0, stream>>>(deg_seq, (size_t)n_op * 2 + n_m);

    // 2) Projections via f32 WMMA; also seed the output accumulators.
    gemm16_wmma_kernel<<<n_op / 16, 256, 0, stream>>>(H_op, W_op, b_op, proj_op, out_op);
    gemm16_wmma_kernel<<<n_m  / 16, 256, 0, stream>>>(H_m,  W_m,  b_m,  proj_m,  out_m);

    // 3) Degrees for the three mean-aggregations.
    degree_kernel<<<(e_seq + 255) / 256, 256, 0, stream>>>(E_seq + e_seq, e_seq, deg_seq);
    degree_kernel<<<(e_o2m + 255) / 256, 256, 0, stream>>>(E_op2m + e_o2m, e_o2m, deg_m);
    degree_kernel<<<(e_o2m + 255) / 256, 256, 0, stream>>>(E_op2m, e_o2m, deg_op);

    // 4) Edge scatter (pre-normalized by clamped degree) into the outputs.
    scatter_seq_kernel<<<e_seq, 128, 0, stream>>>(E_seq, e_seq, proj_op, deg_seq, out_op);
    scatter_op2m_kernel<<<e_o2m, 128, 0, stream>>>(E_op2m, e_o2m, proj_op, proj_m,
                                                   deg_m, deg_op, out_op, out_m);

    // 5) ReLU over the concatenated [op | m] output.
    relu_kernel<<<2048, 256, 0, stream>>>(out_op, (size_t)(n_op + n_m) * 128);
}